// FeatureExtractor_45878840656404
// MI455X (gfx1250) — compile-verified
//
#include <hip/hip_runtime.h>

// ---------------------------------------------------------------------------
// Feature extractor for (B=512, T=8192) fp32 series -> 17 feature planes.
// Bandwidth-bound (~302 MB @ 23.3 TB/s ~= 13us floor). One workgroup per row,
// row staged to LDS via gfx1250 async-to-LDS copies, two-level prefix sums
// (fp64 carries), affine-scan EMAs, non-temporal coalesced output stores.
// ---------------------------------------------------------------------------

#define T_LEN   8192
#define NTH     256              // 8 wave32s
#define SEG     32               // T_LEN / NTH, elements per thread in scan passes
#define NSEG    256              // number of segments (== NTH)
#define BIG     (T_LEN + (T_LEN >> 5))   // +1 float pad per 32 -> bank-conflict-free scans
#define EPSF    1e-8f

struct Smem {
  float  x[BIG];          // staged row
  float  a[BIG];          // local prefix: close           (persistent)
  float  b[BIG];          // local prefix: gain -> |d| -> macd
  float  c[BIG];          // local prefix: loss -> sqdev -> signal
  float  tot[3 * NSEG];   // per-segment totals / affine B terms
  double carD[3 * NSEG];  // fp64 exclusive carries: row0=close, rows1-2 reused
  float  carF[2 * NSEG];  // affine-scan carries for EMA replay
};

__device__ __forceinline__ int pidx(int t) { return t + (t >> 5); }

// Windowed sum with left edge padded by `pad0` (matches reference's repeat-pad).
// P holds segment-local inclusive prefixes; car holds fp64 exclusive segment carries.
__device__ __forceinline__ float wsum(const float* __restrict__ P,
                                      const double* __restrict__ car,
                                      int t, int w, float pad0) {
  double s = car[t >> 5] + (double)P[pidx(t)];
  int u = t - w;
  if (u >= 0) s -= car[u >> 5] + (double)P[pidx(u)];
  float r = (float)s;
  int np = w - 1 - t;
  if (np > 0) r += (float)np * pad0;
  return r;
}

__device__ __forceinline__ void nts(float* p, float v) {
  __builtin_nontemporal_store(v, p);   // streaming store: 94% of traffic is write-once
}

__global__ void __launch_bounds__(NTH, 2)
fe_kernel(const float* __restrict__ close, float* __restrict__ out, int rows) {
  extern __shared__ char smraw[];
  Smem& sm = *reinterpret_cast<Smem*>(smraw);
  const int tid = threadIdx.x;
  const int row = blockIdx.x;
  const float* __restrict__ xg = close + (size_t)row * T_LEN;
  const size_t OS = (size_t)rows * T_LEN;          // stride between feature planes
  float* const ob = out + (size_t)row * T_LEN;     // plane o at ob + o*OS

  // ---------------- Phase 0: stage row (32 KB) into LDS --------------------
  // b32 async packets: each lands exactly at its padded LDS slot (4B aligned),
  // global side fully coalesced (t = k*256 + tid).
#if defined(__AMDGCN__) && __has_builtin(__builtin_amdgcn_global_load_async_to_lds_b32)
  {
    typedef __attribute__((address_space(1))) int* gip_t;
    typedef __attribute__((address_space(3))) int* lip_t;
#pragma unroll
    for (int k = 0; k < SEG; ++k) {
      const int t = k * NTH + tid;
      __builtin_amdgcn_global_load_async_to_lds_b32(
          (gip_t)(xg + t), (lip_t)&sm.x[pidx(t)], 0, 0);
    }
#if __has_builtin(__builtin_amdgcn_s_wait_asynccnt)
    __builtin_amdgcn_s_wait_asynccnt(0);
#else
    asm volatile("s_wait_asynccnt 0" ::: "memory");
#endif
  }
#else
  {
#pragma unroll
    for (int k = 0; k < SEG; ++k) {
      const int t = k * NTH + tid;
      __builtin_prefetch(xg + t, 0, 0);            // global_prefetch_b8 on gfx1250
      sm.x[pidx(t)] = xg[t];
    }
  }
#endif
  __syncthreads();

  const float x0 = sm.x[pidx(0)];

  // ---------------- Phase 1: fused local prefixes: close, gain, loss -------
  {
    const int base = tid * SEG;
    float cc = 0.f, cg = 0.f, cl = 0.f;
    float prev = (base > 0) ? sm.x[pidx(base - 1)] : 0.f;
#pragma unroll
    for (int j = 0; j < SEG; ++j) {
      const int t = base + j;
      const float xv = sm.x[pidx(t)];
      const float d = (t == 0) ? 0.f : (xv - prev);
      prev = xv;
      cc += xv;
      cg += (d > 0.f) ? d : 0.f;
      cl += (d < 0.f) ? -d : 0.f;
      sm.a[pidx(t)] = cc;
      sm.b[pidx(t)] = cg;
      sm.c[pidx(t)] = cl;
    }
    sm.tot[0 * NSEG + tid] = cc;
    sm.tot[1 * NSEG + tid] = cg;
    sm.tot[2 * NSEG + tid] = cl;
  }
  __syncthreads();
  if (tid < 3) {                                    // fp64 exclusive segment carries
    double r = 0.0;
    for (int i = 0; i < NSEG; ++i) { sm.carD[tid * NSEG + i] = r; r += (double)sm.tot[tid * NSEG + i]; }
  }
  __syncthreads();

  // ---------------- Outputs 0..8, 13: MAs, ratios, RSI, BB mid -------------
#pragma unroll 4
  for (int k = 0; k < SEG; ++k) {
    const int t = tid + k * NTH;                    // coalesced
    const float xv   = sm.x[pidx(t)];
    const float ma5  = wsum(sm.a, sm.carD + 0 * NSEG, t, 5,  x0) * (1.f / 5.f);
    const float ma10 = wsum(sm.a, sm.carD + 0 * NSEG, t, 10, x0) * (1.f / 10.f);
    const float ma20 = wsum(sm.a, sm.carD + 0 * NSEG, t, 20, x0) * (1.f / 20.f);
    const float ma50 = wsum(sm.a, sm.carD + 0 * NSEG, t, 50, x0) * (1.f / 50.f);
    nts(ob + 0 * OS + t, ma5);
    nts(ob + 1 * OS + t, xv / (ma5 + EPSF));
    nts(ob + 2 * OS + t, ma10);
    nts(ob + 3 * OS + t, xv / (ma10 + EPSF));
    nts(ob + 4 * OS + t, ma20);
    nts(ob + 5 * OS + t, xv / (ma20 + EPSF));
    nts(ob + 6 * OS + t, ma50);
    nts(ob + 7 * OS + t, xv / (ma50 + EPSF));
    const float ag = wsum(sm.b, sm.carD + 1 * NSEG, t, 14, 0.f) * (1.f / 14.f);
    const float al = wsum(sm.c, sm.carD + 2 * NSEG, t, 14, 0.f) * (1.f / 14.f);
    const float rs = ag / (al + EPSF);
    nts(ob + 8 * OS + t, 100.f - 100.f / (1.f + rs));
    nts(ob + 13 * OS + t, ma20);
  }
  __syncthreads();

  // ---------------- Phase 2: prefixes of |delta| and (x-ma20)^2 ------------
  {
    const int base = tid * SEG;
    float ca = 0.f, cs = 0.f;
    float prev = (base > 0) ? sm.x[pidx(base - 1)] : 0.f;
#pragma unroll
    for (int j = 0; j < SEG; ++j) {
      const int t = base + j;
      const float xv = sm.x[pidx(t)];
      const float d = (t == 0) ? 0.f : (xv - prev);
      prev = xv;
      ca += fabsf(d);
      const float m20 = wsum(sm.a, sm.carD + 0 * NSEG, t, 20, x0) * (1.f / 20.f);
      const float dv = xv - m20;
      cs += dv * dv;
      sm.b[pidx(t)] = ca;                           // reuse (gain prefix consumed)
      sm.c[pidx(t)] = cs;                           // reuse (loss prefix consumed)
    }
    sm.tot[0 * NSEG + tid] = ca;
    sm.tot[1 * NSEG + tid] = cs;
  }
  __syncthreads();
  if (tid < 2) {
    double r = 0.0;
    for (int i = 0; i < NSEG; ++i) { sm.carD[(tid + 1) * NSEG + i] = r; r += (double)sm.tot[tid * NSEG + i]; }
  }
  __syncthreads();

  // ---------------- Outputs 12, 14, 15, 16: Bollinger, %B, ATR -------------
#pragma unroll 4
  for (int k = 0; k < SEG; ++k) {
    const int t = tid + k * NTH;
    const float xv  = sm.x[pidx(t)];
    const float m20 = wsum(sm.a, sm.carD + 0 * NSEG, t, 20, x0) * (1.f / 20.f);
    const float var = wsum(sm.c, sm.carD + 2 * NSEG, t, 20, sm.c[pidx(0)]) * (1.f / 20.f);
    const float atr = wsum(sm.b, sm.carD + 1 * NSEG, t, 14, 0.f) * (1.f / 14.f);
    const float sd  = sqrtf(var + EPSF);
    const float bu  = m20 + 2.f * sd;
    const float bl  = m20 - 2.f * sd;
    nts(ob + 12 * OS + t, bu);
    nts(ob + 14 * OS + t, bl);
    nts(ob + 15 * OS + t, (xv - bl) / (bu - bl + EPSF));
    nts(ob + 16 * OS + t, atr);
  }
  __syncthreads();

  // ---------------- Phase 3: EMA12 / EMA26 via affine-map scan -------------
  const float A12 = 2.f / 13.f, K12 = 1.f - A12;
  const float A26 = 2.f / 27.f, K26 = 1.f - A26;
  {                                                 // segment B terms (replay from 0)
    const int base = tid * SEG;
    float y12 = 0.f, y26 = 0.f;
#pragma unroll
    for (int j = 0; j < SEG; ++j) {
      const int t = base + j;
      const float xv = sm.x[pidx(t)];
      if (t == 0) { y12 = xv; y26 = xv; }
      else { y12 = A12 * xv + K12 * y12; y26 = A26 * xv + K26 * y26; }
    }
    sm.tot[0 * NSEG + tid] = y12;
    sm.tot[1 * NSEG + tid] = y26;
  }
  __syncthreads();
  if (tid == 0) {                                   // serial affine combine (256 steps)
    float m12 = 1.f, m26 = 1.f;
    for (int k = 0; k < SEG; ++k) { m12 *= K12; m26 *= K26; }
    float r12 = 0.f, r26 = 0.f;
    for (int i = 0; i < NSEG; ++i) {
      sm.carF[0 * NSEG + i] = r12;
      sm.carF[1 * NSEG + i] = r26;
      const float M12 = (i == 0) ? 0.f : m12;
      const float M26 = (i == 0) ? 0.f : m26;
      r12 = M12 * r12 + sm.tot[0 * NSEG + i];
      r26 = M26 * r26 + sm.tot[1 * NSEG + i];
    }
  }
  __syncthreads();
  {                                                 // replay: macd -> sm.b
    const int base = tid * SEG;
    float y12 = sm.carF[0 * NSEG + tid];
    float y26 = sm.carF[1 * NSEG + tid];
#pragma unroll
    for (int j = 0; j < SEG; ++j) {
      const int t = base + j;
      const float xv = sm.x[pidx(t)];
      if (t == 0) { y12 = xv; y26 = xv; }
      else { y12 = A12 * xv + K12 * y12; y26 = A26 * xv + K26 * y26; }
      sm.b[pidx(t)] = y12 - y26;
    }
  }
  __syncthreads();

  // ---------------- Phase 4: signal = EMA9(macd) ---------------------------
  const float A9 = 0.2f, K9 = 0.8f;
  {
    const int base = tid * SEG;
    float y9 = 0.f;
#pragma unroll
    for (int j = 0; j < SEG; ++j) {
      const int t = base + j;
      const float mv = sm.b[pidx(t)];
      if (t == 0) y9 = mv; else y9 = A9 * mv + K9 * y9;
    }
    sm.tot[0 * NSEG + tid] = y9;
  }
  __syncthreads();
  if (tid == 0) {
    float m9 = 1.f;
    for (int k = 0; k < SEG; ++k) m9 *= K9;
    float r9 = 0.f;
    for (int i = 0; i < NSEG; ++i) {
      sm.carF[0 * NSEG + i] = r9;
      r9 = ((i == 0) ? 0.f : m9) * r9 + sm.tot[0 * NSEG + i];
    }
  }
  __syncthreads();
  {                                                 // replay: signal -> sm.c
    const int base = tid * SEG;
    float y9 = sm.carF[0 * NSEG + tid];
#pragma unroll
    for (int j = 0; j < SEG; ++j) {
      const int t = base + j;
      const float mv = sm.b[pidx(t)];
      if (t == 0) y9 = mv; else y9 = A9 * mv + K9 * y9;
      sm.c[pidx(t)] = y9;
    }
  }
  __syncthreads();

  // ---------------- Outputs 9, 10, 11: MACD, signal, histogram -------------
#pragma unroll 4
  for (int k = 0; k < SEG; ++k) {
    const int t = tid + k * NTH;
    const float m = sm.b[pidx(t)];
    const float s = sm.c[pidx(t)];
    nts(ob + 9 * OS + t, m);
    nts(ob + 10 * OS + t, s);
    nts(ob + 11 * OS + t, m - s);
  }
}

extern "C" void kernel_launch(void* const* d_in, const int* in_sizes, int n_in,
                              void* d_out, int out_size, void* d_ws, size_t ws_size,
                              hipStream_t stream) {
  (void)n_in; (void)d_ws; (void)ws_size; (void)out_size;
  const float* close = (const float*)d_in[0];      // d_in[1] (volume) unused by reference
  float* out = (float*)d_out;
  const int rows = in_sizes[0] / T_LEN;            // 512
  fe_kernel<<<dim3(rows), dim3(NTH), sizeof(Smem), stream>>>(close, out, rows);
}